// FlexGroupedVarPatchEmbed_18176301597285
// MI455X (gfx1250) — compile-verified
//
#include <hip/hip_runtime.h>

typedef __attribute__((ext_vector_type(2))) float v2f;
typedef __attribute__((ext_vector_type(4))) float v4f;
typedef __attribute__((ext_vector_type(8))) float v8f;

// ---- Problem constants (from reference) ----
constexpr int B_   = 2;
constexpr int V_   = 16;
constexpr int HIMG = 512;
constexpr int WIMG = 512;
constexpr int PP   = 8;              // patch size
constexpr int WW   = WIMG / PP;      // 64 patches per image row
constexpr int L_   = (HIMG / PP) * WW; // 4096 patches
constexpr int K_   = PP * PP;        // 64 = reduction dim
constexpr int E_   = 1024;           // embed dim

// ---- Tiling ----
constexpr int WAVES           = 8;                     // 256 threads
constexpr int ECHUNK          = WAVES * 16;            // 128 embed cols / block
constexpr int NE_CHUNKS       = E_ / ECHUNK;           // 8
constexpr int TILES_L         = L_ / 16;               // 256 patch tiles of 16
constexpr int TILE_GROUPS     = 8;
constexpr int TILES_PER_BLOCK = TILES_L / TILE_GROUPS; // 32

// LDS A-tile row stride (floats). 68 => WMMA A-frag b64 reads hit banks
// 4m + {0,1,2,3} for m=0..15 -> all 64 dwords distinct banks (conflict-free).
constexpr int ASTR = 68;

__global__ __launch_bounds__(256)
void patch_embed_wmma_f32(const float* __restrict__ x,      // [B,V,H,W]
                          const int*   __restrict__ vars,   // [V]
                          const float* __restrict__ w,      // [MAXV,E,1,P,P] = [MAXV,E,64]
                          const float* __restrict__ bias,   // [MAXV,E]
                          float*       __restrict__ out)    // [B,V,L,E]
{
    __shared__ float Alds[16 * ASTR];

    const int tid  = threadIdx.x;
    const int lane = tid & 31;
    const int wave = tid >> 5;

    // Block decode: grid.x = B*V*NE_CHUNKS*TILE_GROUPS
    int blk = blockIdx.x;
    const int tg = blk % TILE_GROUPS; blk /= TILE_GROUPS;
    const int ec = blk % NE_CHUNKS;   blk /= NE_CHUNKS;
    const int v  = blk % V_;          blk /= V_;
    const int b  = blk;

    const int var = vars[v];

    const int n   = lane & 15;   // column (and A-row m) owned by this lane
    const int off = lane >> 4;   // 0/1 -> K sub-pair within WMMA fragment
    const int e   = ec * ECHUNK + wave * 16 + n;

    // ---- B fragments: B[k][n] = w[var][e][k]; loaded once, L2-resident ----
    const float* wp = w + ((size_t)var * E_ + e) * K_;
    v2f bfrag[16];
#pragma unroll
    for (int s = 0; s < 16; ++s)
        bfrag[s] = *(const v2f*)(wp + s * 4 + off * 2);

    const float bval = bias[(size_t)var * E_ + e];

    const float* xbv = x   + ((size_t)(b * V_ + v)) * HIMG * WIMG;
    float*       obv = out + ((size_t)(b * V_ + v)) * L_ * E_;

    // Cooperative A-tile load indexing: wave w loads image row w (128 floats,
    // one float4 per lane, fully coalesced 512B per wave).
    const int arow = wave;            // 0..7 row within patch
    const int acg  = lane;            // float4 group within 128-float row
    const int am   = acg >> 1;        // patch index m (0..15), 2 float4s/patch-row
    const int ak   = arow * PP + (acg & 1) * 4; // k = prow*8 + pcol

    for (int ti = 0; ti < TILES_PER_BLOCK; ++ti) {
        const int t   = tg * TILES_PER_BLOCK + ti;
        const int l0  = t * 16;
        const int hh0 = l0 >> 6;       // l = hh*64 + ww
        const int ww0 = l0 & 63;       // 0,16,32,48
        const int r0  = hh0 * PP;
        const int c0  = ww0 * PP;

        __syncthreads();  // previous iteration's reads done before overwrite
        v4f xa = *(const v4f*)(xbv + (size_t)(r0 + arow) * WIMG + c0 + acg * 4);
        *(v4f*)(&Alds[am * ASTR + ak]) = xa;
        __syncthreads();

        // ---- 16 x V_WMMA_F32_16X16X4_F32 over K=64 ----
        v8f c = {};
#pragma unroll
        for (int s = 0; s < 16; ++s) {
            // A frag: lane m = lane&15, VGPR pair = K {s*4+off*2, +1}
            v2f afrag = *(const v2f*)(&Alds[n * ASTR + s * 4 + off * 2]);
            c = __builtin_amdgcn_wmma_f32_16x16x4_f32(
                    false, afrag, false, bfrag[s],
                    (short)0, c, false, false);
        }

        // ---- bias + store: VGPR r holds D[m = r + off*8][n] ----
        float* orow = obv + (size_t)(l0 + off * 8) * E_ + e;
#pragma unroll
        for (int r = 0; r < 8; ++r)
            orow[(size_t)r * E_] = c[r] + bval;
    }
}

extern "C" void kernel_launch(void* const* d_in, const int* in_sizes, int n_in,
                              void* d_out, int out_size, void* d_ws, size_t ws_size,
                              hipStream_t stream) {
    const float* x    = (const float*)d_in[0];
    const int*   vars = (const int*)  d_in[1];
    const float* w    = (const float*)d_in[2];
    const float* bias = (const float*)d_in[3];
    float*       out  = (float*)d_out;

    dim3 grid(B_ * V_ * NE_CHUNKS * TILE_GROUPS);  // 2048 blocks
    dim3 block(256);
    patch_embed_wmma_f32<<<grid, block, 0, stream>>>(x, vars, w, bias, out);
}